// GCN_30185030156396
// MI455X (gfx1250) — compile-verified
//
#include <hip/hip_runtime.h>
#include <hip/hip_bf16.h>

typedef __attribute__((ext_vector_type(2))) float v2f;
typedef __attribute__((ext_vector_type(8))) float v8f;

#define BN_EPS 1e-5f

// ---------------------------------------------------------------------------
// GEMM: Y[n, FO] = X[n, FI] @ W^T   (W is [FO, FI] row-major, PyTorch layout)
// One wave computes a 16-row stripe across ALL column tiles using
// V_WMMA_F32_16X16X4_F32. A-fragment (16x4 f32): lanes 0-15 hold K=0,1 in
// v[0],v[1]; lanes 16-31 hold K=2,3.  B-fragment (4x16): N striped across
// lanes, K across VGPRs (same register pattern, reading W rows = out chans).
// C/D: 8 VGPRs, lanes 0-15 -> M=vgpr, lanes 16-31 -> M=8+vgpr, N=lane%16.
// ---------------------------------------------------------------------------
template<int FI, int FO>
__global__ void gemm_xWT(const float* __restrict__ X, const float* __restrict__ W,
                         float* __restrict__ Y, int nRows) {
  constexpr int CT = (FO + 15) / 16;                 // column tiles
  const int wave    = threadIdx.x >> 5;
  const int lane    = threadIdx.x & 31;
  const int rowTile = blockIdx.x * (blockDim.x >> 5) + wave;
  if (rowTile * 16 >= nRows) return;                 // wave-uniform exit (EXEC all-1 for WMMA)
  const int half = lane >> 4;
  const int r    = lane & 15;

  const float* ap = X + (size_t)(rowTile * 16 + r) * FI + 2 * half;

  const float* bp[CT];
  bool bv[CT];
#pragma unroll
  for (int ct = 0; ct < CT; ++ct) {
    int wr = ct * 16 + r;
    bv[ct] = (wr < FO);
    bp[ct] = W + (size_t)(bv[ct] ? wr : 0) * FI + 2 * half;
  }

  v8f acc[CT];
#pragma unroll
  for (int ct = 0; ct < CT; ++ct) acc[ct] = (v8f){};

  for (int k = 0; k < FI; k += 4) {
    v2f a;
    a.x = ap[k];
    a.y = ap[k + 1];
#pragma unroll
    for (int ct = 0; ct < CT; ++ct) {
      v2f b;
      b.x = bv[ct] ? bp[ct][k]     : 0.0f;
      b.y = bv[ct] ? bp[ct][k + 1] : 0.0f;
      acc[ct] = __builtin_amdgcn_wmma_f32_16x16x4_f32(
          /*neg_a=*/false, a, /*neg_b=*/false, b,
          /*c_mod=*/(short)0, acc[ct], /*reuse_a=*/false, /*reuse_b=*/false);
    }
  }

#pragma unroll
  for (int ct = 0; ct < CT; ++ct) {
    int col = ct * 16 + r;
    if (col < FO) {
#pragma unroll
      for (int i = 0; i < 8; ++i) {
        int m = rowTile * 16 + i + 8 * half;
        Y[(size_t)m * FO + col] = acc[ct][i];
      }
    }
  }
}

// ---------------------------------------------------------------------------
__global__ void zero_f32(float* __restrict__ p, long long n) {
  long long i = (long long)blockIdx.x * blockDim.x + threadIdx.x;
  long long s = (long long)gridDim.x * blockDim.x;
  for (; i < n; i += s) p[i] = 0.0f;
}

// Edge scatter-add in projected (output) feature space: agg[dst] += proj[src].
// Consecutive threads cover consecutive features of one edge -> coalesced
// reads; f32 atomics resolve in L2 (whole working set is L2-resident).
template<int F>
__global__ void edge_scatter(const float* __restrict__ proj,
                             const int* __restrict__ src,
                             const int* __restrict__ dst,
                             float* __restrict__ agg, long long total) {
  long long i = (long long)blockIdx.x * blockDim.x + threadIdx.x;
  long long s = (long long)gridDim.x * blockDim.x;
  for (; i < total; i += s) {
    int e = (int)(i / F);
    int f = (int)(i % F);
    int sn = src[e];
    int dn = dst[e];
    atomicAdd(&agg[(size_t)dn * F + f], proj[(size_t)sn * F + f]);
  }
}

// z = agg + bias + root  (optionally fused ReLU), in place on z.
template<int C, bool RELU>
__global__ void add_bias_root(float* __restrict__ z, const float* __restrict__ bias,
                              const float* __restrict__ root, long long total) {
  long long i = (long long)blockIdx.x * blockDim.x + threadIdx.x;
  long long s = (long long)gridDim.x * blockDim.x;
  for (; i < total; i += s) {
    int c = (int)(i % C);
    float v = z[i] + bias[c] + root[i];
    if (RELU) v = fmaxf(v, 0.0f);
    z[i] = v;
  }
}

// Per-channel sum / sum-of-squares. blockDim must be a multiple of C.
template<int C>
__global__ void bn_stats(const float* __restrict__ z, float* __restrict__ stats, int nRows) {
  const int rowsPerBlock = blockDim.x / C;
  const int c  = threadIdx.x % C;
  const int rl = threadIdx.x / C;
  if (rl >= rowsPerBlock) return;
  const int stride = gridDim.x * rowsPerBlock;
  float s = 0.0f, ss = 0.0f;
  for (int rrow = blockIdx.x * rowsPerBlock + rl; rrow < nRows; rrow += stride) {
    float v = z[(size_t)rrow * C + c];
    s  += v;
    ss += v * v;
  }
  atomicAdd(&stats[c], s);
  atomicAdd(&stats[C + c], ss);
}

template<int C>
__global__ void bn_finalize(const float* __restrict__ stats,
                            const float* __restrict__ gamma,
                            const float* __restrict__ beta,
                            float* __restrict__ sc, float invN) {
  int c = threadIdx.x;
  if (c >= C) return;
  float mean = stats[c] * invN;
  float var  = stats[C + c] * invN - mean * mean;   // biased variance
  float scale = gamma[c] * rsqrtf(var + BN_EPS);
  sc[c]     = scale;
  sc[C + c] = beta[c] - mean * scale;
}

template<int C, bool RELU>
__global__ void bn_apply(float* __restrict__ z, const float* __restrict__ sc, long long total) {
  long long i = (long long)blockIdx.x * blockDim.x + threadIdx.x;
  long long s = (long long)gridDim.x * blockDim.x;
  for (; i < total; i += s) {
    int c = (int)(i % C);
    float v = z[i] * sc[c] + sc[C + c];
    if (RELU) v = fmaxf(v, 0.0f);
    z[i] = v;
  }
}

// Global mean-pool accumulation (sum + counts via atomics).
__global__ void pool_scatter(const float* __restrict__ h, const int* __restrict__ batch,
                             float* __restrict__ pool, float* __restrict__ cnt,
                             long long total) {
  long long i = (long long)blockIdx.x * blockDim.x + threadIdx.x;
  long long s = (long long)gridDim.x * blockDim.x;
  for (; i < total; i += s) {
    int c = (int)(i % 20);
    int n = (int)(i / 20);
    int g = batch[n];
    atomicAdd(&pool[(size_t)g * 20 + c], h[i]);
    if (c == 0) atomicAdd(&cnt[g], 1.0f);
  }
}

__global__ void final_linear(const float* __restrict__ pool, const float* __restrict__ cnt,
                             const float* __restrict__ Wl, const float* __restrict__ bl,
                             float* __restrict__ out, int total) {
  int i = blockIdx.x * blockDim.x + threadIdx.x;
  if (i >= total) return;
  int g = i / 11;
  int j = i % 11;
  float inv = 1.0f / fmaxf(cnt[g], 1.0f);
  float acc = bl[j];
#pragma unroll
  for (int c = 0; c < 20; ++c)
    acc += (pool[(size_t)g * 20 + c] * inv) * Wl[j * 20 + c];
  out[i] = acc;
}

// ---------------------------------------------------------------------------
extern "C" void kernel_launch(void* const* d_in, const int* in_sizes, int n_in,
                              void* d_out, int out_size, void* d_ws, size_t ws_size,
                              hipStream_t stream) {
  const float* x      = (const float*)d_in[0];
  const int*   edge   = (const int*)d_in[1];      // [2, E] flat; row0=src, row1=dst
  const int*   batch  = (const int*)d_in[2];
  const float* W1_rel = (const float*)d_in[3];
  const float* b1     = (const float*)d_in[4];
  const float* W1_root= (const float*)d_in[5];
  const float* W2_rel = (const float*)d_in[6];
  const float* b2     = (const float*)d_in[7];
  const float* W2_root= (const float*)d_in[8];
  const float* W3_rel = (const float*)d_in[9];
  const float* b3     = (const float*)d_in[10];
  const float* W3_root= (const float*)d_in[11];
  const float* gamma1 = (const float*)d_in[12];
  const float* beta1  = (const float*)d_in[13];
  const float* gamma2 = (const float*)d_in[14];
  const float* beta2  = (const float*)d_in[15];
  const float* W_lin  = (const float*)d_in[16];
  const float* b_lin  = (const float*)d_in[17];
  float* out = (float*)d_out;

  const int N = in_sizes[0] / 128;
  const int E = in_sizes[1] / 2;
  const int G = 256;
  const int* src = edge;
  const int* dst = edge + E;

  // Workspace layout (floats): 3 big N*64 buffers + small tail.
  float* ws    = (float*)d_ws;
  float* bufA  = ws;                          // proj  (rel-projected features)
  float* bufB  = ws + (size_t)N * 64;         // root  (root-projected features)
  float* bufC  = ws + (size_t)2 * N * 64;     // agg -> h (in place per layer)
  float* stats = ws + (size_t)3 * N * 64;     // [2*64]
  float* sc    = stats + 128;                 // [2*64] scale/shift
  float* pool  = sc + 128;                    // [G*20]
  float* cnt   = pool + (size_t)G * 20;       // [G]

  const int rowTiles  = (N + 15) / 16;
  const int gemmBlk   = 128;                               // 4 waves
  const int gemmGrid  = (rowTiles + 3) / 4;
  const float invN    = 1.0f / (float)N;

  // ---------------- Layer 1: 128 -> 64, BN + ReLU ----------------
  gemm_xWT<128, 64><<<gemmGrid, gemmBlk, 0, stream>>>(x, W1_rel,  bufA, N);
  gemm_xWT<128, 64><<<gemmGrid, gemmBlk, 0, stream>>>(x, W1_root, bufB, N);
  zero_f32<<<2048, 256, 0, stream>>>(bufC, (long long)N * 64);
  edge_scatter<64><<<8192, 256, 0, stream>>>(bufA, src, dst, bufC, (long long)E * 64);
  add_bias_root<64, false><<<4096, 256, 0, stream>>>(bufC, b1, bufB, (long long)N * 64);
  zero_f32<<<1, 128, 0, stream>>>(stats, 128);
  bn_stats<64><<<256, 256, 0, stream>>>(bufC, stats, N);
  bn_finalize<64><<<1, 64, 0, stream>>>(stats, gamma1, beta1, sc, invN);
  bn_apply<64, true><<<4096, 256, 0, stream>>>(bufC, sc, (long long)N * 64);   // h1 in bufC

  // ---------------- Layer 2: 64 -> 32, ReLU ----------------
  gemm_xWT<64, 32><<<gemmGrid, gemmBlk, 0, stream>>>(bufC, W2_rel,  bufA, N);
  gemm_xWT<64, 32><<<gemmGrid, gemmBlk, 0, stream>>>(bufC, W2_root, bufB, N);
  zero_f32<<<2048, 256, 0, stream>>>(bufC, (long long)N * 32);
  edge_scatter<32><<<8192, 256, 0, stream>>>(bufA, src, dst, bufC, (long long)E * 32);
  add_bias_root<32, true><<<4096, 256, 0, stream>>>(bufC, b2, bufB, (long long)N * 32); // h2

  // ---------------- Layer 3: 32 -> 20, BN (no ReLU) ----------------
  gemm_xWT<32, 20><<<gemmGrid, gemmBlk, 0, stream>>>(bufC, W3_rel,  bufA, N);
  gemm_xWT<32, 20><<<gemmGrid, gemmBlk, 0, stream>>>(bufC, W3_root, bufB, N);
  zero_f32<<<2048, 256, 0, stream>>>(bufC, (long long)N * 20);
  edge_scatter<20><<<8192, 256, 0, stream>>>(bufA, src, dst, bufC, (long long)E * 20);
  add_bias_root<20, false><<<4096, 256, 0, stream>>>(bufC, b3, bufB, (long long)N * 20);
  zero_f32<<<1, 64, 0, stream>>>(stats, 40);
  bn_stats<20><<<256, 320, 0, stream>>>(bufC, stats, N);
  bn_finalize<20><<<1, 32, 0, stream>>>(stats, gamma2, beta2, sc, invN);
  bn_apply<20, false><<<4096, 256, 0, stream>>>(bufC, sc, (long long)N * 20);  // h3

  // ---------------- Global mean pool + final linear ----------------
  zero_f32<<<32, 256, 0, stream>>>(pool, (long long)G * 21);  // pool + cnt (contiguous)
  pool_scatter<<<4096, 256, 0, stream>>>(bufC, batch, pool, cnt, (long long)N * 20);
  final_linear<<<(G * 11 + 255) / 256, 256, 0, stream>>>(pool, cnt, W_lin, b_lin, out, G * 11);
}